// YOLO_V1_Loss_51994874085625
// MI455X (gfx1250) — compile-verified
//
#include <hip/hip_runtime.h>
#include <hip/hip_bf16.h>
#include <math.h>

// ---------------------------------------------------------------------------
// YOLO-v1 loss on MI455X (gfx1250, wave32).
//
// Roofline: 193 MB streamed once @ 23.3 TB/s => ~8.3us floor; ~160 MFLOP of
// VALU => pure bandwidth problem. Strategy:
//   kernel 1: grid-stride streaming, float2 (b64) loads of the 120B AoS
//             records, 6 register accumulators, per-wave reduction via
//             V_WMMA_F32_16X16X4_F32 (A = data in documented 16x4 layout,
//             B = ones => D[m][n] = x_m + x_{m+16}), LDS fold across the
//             8 waves, 6 partials per block into d_ws.
//   kernel 2: one wave deterministically folds all block partials (same WMMA
//             reduction) and applies the 1/batch_size scaling.
// ---------------------------------------------------------------------------

typedef __attribute__((ext_vector_type(2))) float v2f;
typedef __attribute__((ext_vector_type(8))) float v8f;

// Full 32-lane sum, result valid in every lane. Uses the f32 WMMA:
//   A[m][0]=x_m (lanes 0-15 vgpr0), A[m][2]=x_{m+16} (lanes 16-31 vgpr0),
//   A[m][1]=A[m][3]=0, B = all ones (layout-independent).
//   => D[m][n] = x_m + x_{m+16} for every n.
//   Lane L holds D rows {0..7} (L<16) or {8..15} (L>=16) in d[0..7],
//   so sum(d[0..7]) gives half the rowsums; one shfl_xor(16) completes it.
__device__ __forceinline__ float wave_sum32(float x) {
#if defined(__gfx1250__) && __has_builtin(__builtin_amdgcn_wmma_f32_16x16x4_f32)
  v2f a; a[0] = x;    a[1] = 0.0f;
  v2f b; b[0] = 1.0f; b[1] = 1.0f;
  v8f c = {};
  v8f d = __builtin_amdgcn_wmma_f32_16x16x4_f32(false, a, false, b,
                                                (short)0, c, false, false);
  float s = ((d[0] + d[1]) + (d[2] + d[3])) + ((d[4] + d[5]) + (d[6] + d[7]));
  s += __shfl_xor(s, 16, 32);
  return s;
#else
  for (int off = 16; off > 0; off >>= 1) x += __shfl_xor(x, off, 32);
  return x;
#endif
}

__device__ __forceinline__ float iou_box(float bx, float by, float bw, float bh,
                                         float gx, float gy,
                                         float gx1, float gy1, float gx2,
                                         float gy2, float gArea) {
  float cx = truncf(gx + bx * 64.0f);
  float cy = truncf(gy + by * 64.0f);
  float w  = truncf(bw * 448.0f);
  float h  = truncf(bh * 448.0f);
  float x1 = fmaxf(0.0f,   cx - w * 0.5f);
  float y1 = fmaxf(0.0f,   cy - h * 0.5f);
  float x2 = fminf(447.0f, cx + w * 0.5f);
  float y2 = fminf(447.0f, cy + h * 0.5f);
  float pArea = (x2 - x1) * (y2 - y1);
  float lx = fmaxf(x1, gx1);
  float rx = fminf(x2, gx2);
  float uy = fmaxf(y1, gy1);
  float dy = fminf(y2, gy2);
  float inter = (rx - lx) * (dy - uy);
  float denom = pArea + gArea - inter;
  denom = (denom == 0.0f) ? 1.0f : denom;
  float r = inter / denom;
  return ((rx >= lx) && (dy >= uy)) ? r : 0.0f;
}

__global__ __launch_bounds__(256) void yolo_v1_partial(
    const float* __restrict__ pred, const float* __restrict__ gt,
    float* __restrict__ partials, int ncells) {
  const int tid    = blockIdx.x * 256 + threadIdx.x;
  const int stride = gridDim.x * 256;

  float a0 = 0.f, a1 = 0.f, a2 = 0.f, a3 = 0.f, a4 = 0.f, a5 = 0.f;

  for (int c = tid; c < ncells; c += stride) {
    const float* p = pred + (size_t)c * 30;
    const float* g = gt   + (size_t)c * 30;

    // NT prefetch of the next grid-stride record (data is touched once).
    int cn = c + stride;
    if (cn < ncells) {
      __builtin_prefetch(pred + (size_t)cn * 30, 0, 0);
      __builtin_prefetch(gt   + (size_t)cn * 30, 0, 0);
    }

    // Head: boxes + obj/rel fields, b64-vectorized (base is 8B aligned).
    float pv[10], gv[10];
    const float2* ph = (const float2*)p;
    const float2* gh = (const float2*)g;
#pragma unroll
    for (int k = 0; k < 5; ++k) {
      float2 t = ph[k]; pv[2 * k] = t.x; pv[2 * k + 1] = t.y;
    }
#pragma unroll
    for (int k = 0; k < 5; ++k) {
      float2 t = gh[k]; gv[2 * k] = t.x; gv[2 * k + 1] = t.y;
    }

    // Class MSE streamed in float2 pairs (offset 10 floats keeps 8B align).
    const float2* pc = (const float2*)(p + 10);
    const float2* gc = (const float2*)(g + 10);
    float cls = 0.f;
#pragma unroll
    for (int k = 0; k < 10; ++k) {
      float2 pp = pc[k];
      float2 gg = gc[k];
      float t0 = gg.x - pp.x, t1 = gg.y - pp.y;
      cls += t0 * t0 + t1 * t1;
    }
    cls *= 0.05f;  // mean over C=20

    const int within = c % 49;           // cell within image: (row, col)
    const float gyv = (float)(within / 7) * 64.0f;
    const float gxv = (float)(within % 7) * 64.0f;

    const float gx1 = gv[5], gy1 = gv[6], gx2 = gv[7], gy2 = gv[8];
    const float gArea = (gx2 - gx1) * (gy2 - gy1);

    const float iou1 = iou_box(pv[0], pv[1], pv[2], pv[3], gxv, gyv,
                               gx1, gy1, gx2, gy2, gArea);
    const float iou2 = iou_box(pv[5], pv[6], pv[7], pv[8], gxv, gyv,
                               gx1, gy1, gx2, gy2, gArea);
    const bool use1 = iou1 > iou2;

    const float b0 = use1 ? pv[0] : pv[5];
    const float b1 = use1 ? pv[1] : pv[6];
    const float b2 = use1 ? pv[2] : pv[7];
    const float b3 = use1 ? pv[3] : pv[8];
    const float b4 = use1 ? pv[4] : pv[9];
    const float nconf = use1 ? pv[9] : pv[4];
    const float biou  = use1 ? iou1 : iou2;

    const bool obj = (rintf(gv[9]) != 0.0f);

    const float d0 = gv[0] - b0;
    const float d1 = gv[1] - b1;
    const float sw = sqrtf(gv[2] + 1e-8f) - sqrtf(b2 + 1e-8f);
    const float sh = sqrtf(gv[3] + 1e-8f) - sqrtf(b3 + 1e-8f);
    const float coord = 5.0f * (d0 * d0 + d1 * d1 + sw * sw + sh * sh);

    float chd = b4 - biou;
    const float conf_hit = chd * chd;
    const float n2 = nconf * nconf;
    const float noobj = 0.5f * pv[4] * pv[4] + pv[9] * pv[9];

    if (obj) {
      a0 += coord + conf_hit + 0.5f * n2 + cls;
      a1 += coord;
      a2 += conf_hit + n2;
      a3 += cls;
      a4 += biou;
      a5 += 1.0f;
    } else {
      a0 += noobj;
      a2 += noobj;
    }
  }

  // Per-wave WMMA reduction, then LDS fold across the 8 waves of the block.
  __shared__ float smem[8][6];
  const int lane = threadIdx.x & 31;
  const int wv   = threadIdx.x >> 5;
  float r[6] = {a0, a1, a2, a3, a4, a5};
#pragma unroll
  for (int j = 0; j < 6; ++j) {
    float s = wave_sum32(r[j]);
    if (lane == 0) smem[wv][j] = s;
  }
  __syncthreads();
  if (threadIdx.x < 6) {
    float t = 0.f;
#pragma unroll
    for (int w = 0; w < 8; ++w) t += smem[w][threadIdx.x];
    partials[blockIdx.x * 6 + threadIdx.x] = t;
  }
}

__global__ __launch_bounds__(32) void yolo_v1_finalize(
    const float* __restrict__ partials, int nparts,
    const int* __restrict__ bsp, float* __restrict__ out) {
  const int lane = threadIdx.x;  // one full wave32
  float acc[6] = {0.f, 0.f, 0.f, 0.f, 0.f, 0.f};
  for (int i = lane; i < nparts; i += 32) {
#pragma unroll
    for (int j = 0; j < 6; ++j) acc[j] += partials[i * 6 + j];
  }
  float tot[6];
#pragma unroll
  for (int j = 0; j < 6; ++j) tot[j] = wave_sum32(acc[j]);
  if (lane == 0) {
    const float inv = 1.0f / (float)(*bsp);
    out[0] = tot[0] * inv;  // loss / bs
    out[1] = tot[1] * inv;  // loss_coord / bs
    out[2] = tot[2] * inv;  // loss_conf / bs
    out[3] = tot[3] * inv;  // loss_cls / bs
    out[4] = tot[4];        // iou_sum
    out[5] = tot[5];        // obj_num
  }
}

extern "C" void kernel_launch(void* const* d_in, const int* in_sizes, int n_in,
                              void* d_out, int out_size, void* d_ws,
                              size_t ws_size, hipStream_t stream) {
  const float* pred = (const float*)d_in[0];
  const float* gt   = (const float*)d_in[1];
  const int*   bsp  = (const int*)d_in[2];
  float* out      = (float*)d_out;
  float* partials = (float*)d_ws;

  const int ncells = in_sizes[0] / 30;  // N*S*S

  int nblocks = 1024;  // ~262k threads; enough waves to saturate HBM
  const size_t need = (size_t)nblocks * 6 * sizeof(float);
  if (ws_size < need) {
    nblocks = (int)(ws_size / (6 * sizeof(float)));
    if (nblocks < 1) nblocks = 1;
  }

  yolo_v1_partial<<<nblocks, 256, 0, stream>>>(pred, gt, partials, ncells);
  yolo_v1_finalize<<<1, 32, 0, stream>>>(partials, nblocks, bsp, out);
}